// Digits_Caps_22033182228900
// MI455X (gfx1250) — compile-verified
//
#include <hip/hip_runtime.h>
#include <stdint.h>

// Problem constants
#define BATCH 32
#define IUNITS 16
#define CCH 8192
#define UU 32
#define KK 16
#define SLOTS 512          // (b,k) slots per (c,u), WMMA-permuted
#define HOT_C 5120         // u_hat channels kept L2-resident (160 MB of 192 MB L2); multiple of 512

typedef __attribute__((ext_vector_type(2))) float v2f;
typedef __attribute__((ext_vector_type(4))) float v4f;
typedef __attribute__((ext_vector_type(8))) float v8f;
typedef __attribute__((ext_vector_type(4))) unsigned v4u;

// round-to-nearest + single v_perm_b32 pack of two bf16
__device__ __forceinline__ unsigned pk_bf16(float lo, float hi) {
    union { float f; unsigned u; } a, b; a.f = lo; b.f = hi;
    unsigned au = a.u + 0x8000u, bu = b.u + 0x8000u;
#if __has_builtin(__builtin_amdgcn_perm)
    return __builtin_amdgcn_perm(bu, au, 0x07060302u);  // {bu[31:16], au[31:16]}
#else
    return (bu & 0xFFFF0000u) | (au >> 16);
#endif
}
__device__ __forceinline__ float bfhi2f(unsigned hibits) { // bf16 already in [31:16]
    union { unsigned u; float f; } x; x.u = hibits; return x.f;
}

// ---------------------------------------------------------------------------
// Kernel 0: transpose x -> xT[c][b][i], zero b_ij
// ---------------------------------------------------------------------------
__global__ __launch_bounds__(256) void k_init(const float* __restrict__ x,
                                              float* __restrict__ xT,
                                              float* __restrict__ bij) {
    int t = blockIdx.x * 256 + threadIdx.x;        // 1M threads, 4 elems each
    float4 r = ((const float4*)x)[t];              // idx = (b*16+i)*8192 + c, c fast
    int idx0 = t * 4;
    int bi = idx0 >> 13;                           // b*16+i
    int b = bi >> 4, i = bi & 15;
    int c0 = idx0 & (CCH - 1);
    xT[((size_t)(c0 + 0) * BATCH + b) * IUNITS + i] = r.x;
    xT[((size_t)(c0 + 1) * BATCH + b) * IUNITS + i] = r.y;
    xT[((size_t)(c0 + 2) * BATCH + b) * IUNITS + i] = r.z;
    xT[((size_t)(c0 + 3) * BATCH + b) * IUNITS + i] = r.w;
    if (t < CCH * UU) bij[t] = 0.0f;
}

// ---------------------------------------------------------------------------
// Kernel 1: u_hat via V_WMMA_F32_16X16X4_F32, one channel per block.
// Stores bf16, slot = h*256 + lane*8 + r (WMMA C/D layout).
// W streamed with NT loads; cold-region u_hat stored NT to protect hot L2 set.
// ---------------------------------------------------------------------------
__global__ __launch_bounds__(256) void k_uhat(const float* __restrict__ xT,
                                              const float* __restrict__ W,
                                              unsigned* __restrict__ uh32) {
    __shared__ __align__(16) float lx[BATCH * IUNITS];     // 2 KB
    __shared__ __align__(16) float lw[UU * KK * IUNITS];   // 32 KB
    int c = blockIdx.x;
    int tid = threadIdx.x;

    const v4f* Wc = (const v4f*)(W + (size_t)c * (UU * KK * IUNITS));
    v4f* lw4 = (v4f*)lw;
    for (int j = tid; j < (UU * KK * IUNITS) / 4; j += 256)
        lw4[j] = __builtin_nontemporal_load(Wc + j);
    if (tid < (BATCH * IUNITS) / 4)
        ((v4f*)lx)[tid] = ((const v4f*)(xT + (size_t)c * BATCH * IUNITS))[tid];
    __syncthreads();

    int wave = tid >> 5, lane = tid & 31;
    int m = lane & 15;                 // A row (b-local) == B col (k)
    int koff = (lane >> 4) << 1;       // lanes 16-31 hold K+2 / K+3
    bool hot = (c < HOT_C);

    for (int tu = 0; tu < 4; ++tu) {
        int u = wave * 4 + tu;
        for (int h = 0; h < 2; ++h) {
            v8f acc = {};
#pragma unroll
            for (int step = 0; step < 4; ++step) {
                int ib = step * 4 + koff;
                v2f a, b;
                a.x = lx[(h * 16 + m) * 16 + ib];
                a.y = lx[(h * 16 + m) * 16 + ib + 1];
                b.x = lw[u * 256 + m * 16 + ib];
                b.y = lw[u * 256 + m * 16 + ib + 1];
                acc = __builtin_amdgcn_wmma_f32_16x16x4_f32(
                    false, a, false, b, (short)0, acc, false, false);
            }
            v4u o;
            o.x = pk_bf16(acc[0], acc[1]);
            o.y = pk_bf16(acc[2], acc[3]);
            o.z = pk_bf16(acc[4], acc[5]);
            o.w = pk_bf16(acc[6], acc[7]);
            v4u* dst = (v4u*)uh32 + (((size_t)c * UU + u) * 2 + h) * 32 + lane;
            if (hot) *dst = o;
            else     __builtin_nontemporal_store(o, dst);
        }
    }
}

// ---------------------------------------------------------------------------
// Kernel 2: softmax over channels, one u per block
// ---------------------------------------------------------------------------
__global__ __launch_bounds__(256) void k_softmax(const float* __restrict__ bij,
                                                 float* __restrict__ cij) {
    __shared__ float red[256];
    int u = blockIdx.x, tid = threadIdx.x;
    const float* row = bij + (size_t)u * CCH;
    float m = -3.4e38f;
    for (int c = tid; c < CCH; c += 256) m = fmaxf(m, row[c]);
    red[tid] = m; __syncthreads();
    for (int s = 128; s > 0; s >>= 1) { if (tid < s) red[tid] = fmaxf(red[tid], red[tid + s]); __syncthreads(); }
    float mx = red[0]; __syncthreads();
    float sum = 0.0f;
    for (int c = tid; c < CCH; c += 256) sum += __expf(row[c] - mx);
    red[tid] = sum; __syncthreads();
    for (int s = 128; s > 0; s >>= 1) { if (tid < s) red[tid] += red[tid + s]; __syncthreads(); }
    float inv = 1.0f / red[0];
    float* out = cij + (size_t)u * CCH;
    for (int c = tid; c < CCH; c += 256) out[c] = __expf(row[c] - mx) * inv;
}

__global__ void k_zero_s(float* __restrict__ s) {
    int g = blockIdx.x * 256 + threadIdx.x;
    ((float4*)s)[g] = float4{0.f, 0.f, 0.f, 0.f};
}

// ---------------------------------------------------------------------------
// Kernel 3: s[u][slot] = sum_c c_ij[u][c] * u_hat[c][u][slot]
// grid = (C/512, U); hot chunks use RT loads (L2), cold chunks NT (HBM stream)
// ---------------------------------------------------------------------------
__global__ __launch_bounds__(256) void k_s(const unsigned* __restrict__ uh32,
                                           const float* __restrict__ cij,
                                           float* __restrict__ s, int uniform) {
    __shared__ float lw[512];
    int u = blockIdx.y, c0 = blockIdx.x * 512, tid = threadIdx.x;
    if (uniform) {
        lw[tid] = 1.0f / CCH; lw[tid + 256] = 1.0f / CCH;
    } else {
        lw[tid]       = cij[(size_t)u * CCH + c0 + tid];
        lw[tid + 256] = cij[(size_t)u * CCH + c0 + tid + 256];
    }
    __syncthreads();
    float a0 = 0.f, a1 = 0.f;
    if (c0 < HOT_C) {
#pragma unroll 4
        for (int cc = 0; cc < 512; ++cc) {
            unsigned p = uh32[((size_t)(c0 + cc) * UU + u) * 256 + tid];
            float w = lw[cc];
            a0 += w * bfhi2f(p << 16);
            a1 += w * bfhi2f(p & 0xFFFF0000u);
        }
    } else {
#pragma unroll 4
        for (int cc = 0; cc < 512; ++cc) {
            unsigned p = __builtin_nontemporal_load(
                &uh32[((size_t)(c0 + cc) * UU + u) * 256 + tid]);
            float w = lw[cc];
            a0 += w * bfhi2f(p << 16);
            a1 += w * bfhi2f(p & 0xFFFF0000u);
        }
    }
    atomicAdd(&s[u * SLOTS + tid * 2],     a0);
    atomicAdd(&s[u * SLOTS + tid * 2 + 1], a1);
}

// ---------------------------------------------------------------------------
// Kernel 4: squash (reduction over u is slot-aligned across all u)
// ---------------------------------------------------------------------------
__global__ void k_squash(const float* __restrict__ s, float* __restrict__ v) {
    int slot = blockIdx.x * 256 + threadIdx.x;     // 512 total
    float msq = 0.f;
    for (int u = 0; u < UU; ++u) { float t = s[u * SLOTS + slot]; msq += t * t; }
    float scale = msq / ((1.0f + msq) * sqrtf(msq));
    for (int u = 0; u < UU; ++u) v[u * SLOTS + slot] = s[u * SLOTS + slot] * scale;
}

// ---------------------------------------------------------------------------
// Kernel 5: b_ij[u][c] += dot(u_hat[c][u][:], v[u][:]) / B  (per-channel, no
// global reduction). One wave per channel, wave32 shuffle reduction.
// ---------------------------------------------------------------------------
__global__ __launch_bounds__(256) void k_a(const unsigned* __restrict__ uh32,
                                           const float* __restrict__ v,
                                           float* __restrict__ bij) {
    __shared__ float lv[512];
    int u = blockIdx.y, c0 = blockIdx.x * 512, tid = threadIdx.x;
    lv[tid]       = v[u * SLOTS + tid];
    lv[tid + 256] = v[u * SLOTS + tid + 256];
    __syncthreads();
    int wave = tid >> 5, lane = tid & 31;
    const v4u* uh16 = (const v4u*)uh32;
    bool hot = (c0 < HOT_C);
    for (int j = 0; j < 64; ++j) {
        int c = c0 + wave * 64 + j;
        size_t base = ((size_t)c * UU + u) * 64;           // v4u units per (c,u)
        v4u q0, q1;
        if (hot) {
            q0 = uh16[base + lane * 2];
            q1 = uh16[base + lane * 2 + 1];
        } else {
            q0 = __builtin_nontemporal_load(&uh16[base + lane * 2]);
            q1 = __builtin_nontemporal_load(&uh16[base + lane * 2 + 1]);
        }
        const float* vv = &lv[lane * 16];
        unsigned w[8] = {q0.x, q0.y, q0.z, q0.w, q1.x, q1.y, q1.z, q1.w};
        float sum = 0.f;
#pragma unroll
        for (int p = 0; p < 8; ++p) {
            sum += vv[p * 2]     * bfhi2f(w[p] << 16);
            sum += vv[p * 2 + 1] * bfhi2f(w[p] & 0xFFFF0000u);
        }
#pragma unroll
        for (int off = 16; off > 0; off >>= 1) sum += __shfl_xor(sum, off, 32);
        if (lane == 0) bij[(size_t)u * CCH + c] += sum * (1.0f / BATCH);
    }
}

// ---------------------------------------------------------------------------
// Kernel 6: un-permute slots -> out[b][u][k]
// ---------------------------------------------------------------------------
__global__ void k_out(const float* __restrict__ v, float* __restrict__ out) {
    int o = blockIdx.x * 256 + threadIdx.x;        // 16384
    int k = o & 15, u = (o >> 4) & 31, b = o >> 9;
    int h = b >> 4, bl = b & 15;
    int lane = (bl < 8) ? k : (k + 16);
    int r    = (bl < 8) ? bl : (bl - 8);
    int slot = h * 256 + lane * 8 + r;
    out[o] = v[u * SLOTS + slot];
}

// ---------------------------------------------------------------------------
extern "C" void kernel_launch(void* const* d_in, const int* in_sizes, int n_in,
                              void* d_out, int out_size, void* d_ws, size_t ws_size,
                              hipStream_t stream) {
    const float* x = (const float*)d_in[0];     // (32,16,8192) f32
    const float* W = (const float*)d_in[1];     // (8192,32,16,16) f32
    float* out = (float*)d_out;                 // 16384 f32

    char* ws = (char*)d_ws;
    unsigned* uh = (unsigned*)ws;                                  // 268435456 B
    float* xT  = (float*)(ws + 268435456ull);                      // 16777216 B
    float* bij = (float*)(ws + 268435456ull + 16777216ull);        // 1 MB
    float* cij = bij + (size_t)CCH * UU;                           // 1 MB
    float* s   = cij + (size_t)CCH * UU;                           // 64 KB
    float* v   = s + (size_t)UU * SLOTS;                           // 64 KB

    k_init<<<4096, 256, 0, stream>>>(x, xT, bij);
    k_uhat<<<CCH, 256, 0, stream>>>(xT, W, uh);
    for (int t = 0; t < 3; ++t) {
        if (t > 0) k_softmax<<<UU, 256, 0, stream>>>(bij, cij);
        k_zero_s<<<16, 256, 0, stream>>>(s);
        k_s<<<dim3(16, 32), 256, 0, stream>>>(uh, cij, s, (t == 0) ? 1 : 0);
        k_squash<<<2, 256, 0, stream>>>(s, v);
        if (t < 2) k_a<<<dim3(16, 32), 256, 0, stream>>>(uh, v, bij);  // last a never affects v
    }
    k_out<<<64, 256, 0, stream>>>(v, out);
}